// ILA_Layer_36627481100855
// MI455X (gfx1250) — compile-verified
//
#include <hip/hip_runtime.h>
#include <stdint.h>

// MI455X / gfx1250 ILA local-attention layer.
// Stage 1: key/query projections as FP32 WMMA GEMMs (v_wmma_f32_16x16x4_f32).
// Stage 2: per-row attention:
//   - async global->LDS staging of padded key/ft rows (ASYNCcnt path)
//   - sim   = banded WMMA GEMM (band of Q^T x K_row)
//   - softmax(81)
//   - out   = banded WMMA GEMM (ft_row x banded-weight matrix)

typedef float v2f __attribute__((ext_vector_type(2)));
typedef float v8f __attribute__((ext_vector_type(8)));

#define CCH   128          // channels
#define HH    64
#define WW    64
#define PP    (HH * WW)    // 4096 pixels per image
#define LWIN  9
#define KWIN  81
#define PADW  4
#define WCOLS (WW + 2 * PADW)  // 72 padded columns
#define QP    72           // LDS pitch: 2*72 % 64 == 16 -> half-wave rows on disjoint banks
#define RP    72
#define SP    81
#define WBP   72

// ---------------------------------------------------------------------------
// Stage 1: Y[n,o,p] = sum_i W[o,i] * X[n,i,p]   (1x1 conv == channel GEMM)
// ---------------------------------------------------------------------------
__global__ __launch_bounds__(256) void ila_proj_gemm_f32(
    const float* __restrict__ Wm, const float* __restrict__ ft,
    const float* __restrict__ fk, float* __restrict__ key,
    float* __restrict__ query) {
  const int lane = threadIdx.x & 31;
  const int wv   = threadIdx.x >> 5;   // 8 waves -> 8 o-tiles = all 128 channels
  const int lm   = lane & 15;
  const int hi   = lane >> 4;
  const int ks   = hi * 2;             // K sub-offset per ISA f32 A/B layout
  const int o0   = wv * 16;
  const int p0   = blockIdx.x * 16;
  const int n    = blockIdx.y;

  const float* X = (blockIdx.z == 0) ? ft : fk;
  float*       Y = (blockIdx.z == 0) ? key : query;
  const float* Xn = X + (size_t)n * CCH * PP;
  float*       Yn = Y + (size_t)n * CCH * PP;

  v8f acc = {};
#pragma unroll
  for (int kk = 0; kk < CCH / 4; ++kk) {
    const int k = kk * 4;
    v2f a, b;
    a.x = Wm[(o0 + lm) * CCH + k + ks];
    a.y = Wm[(o0 + lm) * CCH + k + ks + 1];
    b.x = Xn[(size_t)(k + ks) * PP + p0 + lm];
    b.y = Xn[(size_t)(k + ks + 1) * PP + p0 + lm];
    acc = __builtin_amdgcn_wmma_f32_16x16x4_f32(false, a, false, b, (short)0,
                                                acc, false, false);
  }
#pragma unroll
  for (int r = 0; r < 8; ++r) {
    const int o = o0 + r + hi * 8;
    Yn[(size_t)o * PP + p0 + lm] = acc[r];
  }
}

// ---------------------------------------------------------------------------
// Async staging helpers (gfx1250 ASYNCcnt path)
// ---------------------------------------------------------------------------
__device__ __forceinline__ void async_wait0() {
  asm volatile("s_wait_asynccnt 0x0" ::: "memory");
}

// Issue async copies of padded row hr of src (layout (C,H,W) at nbase) into
// rb[c*RP + 4 + w]; zero the 8 pad columns (or whole buffer if row OOB).
// hr is uniform across the block, so branches are wave-uniform.
__device__ __forceinline__ void stage_row_issue(const float* __restrict__ src,
                                                size_t nbase, int hr,
                                                float* rb, int tid) {
  if ((unsigned)hr < HH) {
    // zero left/right pad columns (8 per channel)
    for (int i = tid; i < CCH * 8; i += 256) {
      const int c = i >> 3, j = i & 7;
      rb[c * RP + (j < PADW ? j : j + WW)] = 0.0f;
    }
    // async 16B copies: 128 channels x 16 chunks, direct global -> LDS
    const float* rowp = src + nbase + (size_t)hr * WW;
    for (int i = tid; i < CCH * (WW / 4); i += 256) {
      const int c = i >> 4, wq = (i & 15) * 4;
      const unsigned ldst = (unsigned)(uintptr_t)&rb[c * RP + PADW + wq];
      const float* g = rowp + (size_t)c * PP + wq;
      asm volatile("global_load_async_to_lds_b128 %0, %1, off" ::"v"(ldst),
                   "v"(g)
                   : "memory");
    }
  } else {
    for (int i = tid; i < CCH * WCOLS; i += 256)
      rb[(i / WCOLS) * RP + (i % WCOLS)] = 0.0f;
  }
}

// ---------------------------------------------------------------------------
// Stage 2: one workgroup per (n, h) output row, 256 threads (8 waves).
// ---------------------------------------------------------------------------
__global__ __launch_bounds__(256) void ila_attn_f32(
    const float* __restrict__ ft, const float* __restrict__ key,
    const float* __restrict__ query, float* __restrict__ out) {
  extern __shared__ float smem[];
  float* qs  = smem;               // CCH * QP   : query row
  float* rb  = qs + CCH * QP;      // CCH * RP   : padded key/ft row (shared)
  float* sim = rb + CCH * RP;      // WW  * SP   : sim, then softmax weights
  float* wb  = sim + WW * SP;      // WCOLS * WBP: banded weight matrix

  const int tid  = threadIdx.x;
  const int lane = tid & 31;
  const int wv   = tid >> 5;
  const int lm   = lane & 15;
  const int hi   = lane >> 4;
  const int ks   = hi * 2;
  const int h    = blockIdx.x;
  const int n    = blockIdx.y;
  const size_t nbase = (size_t)n * CCH * PP;

  // phase 0: async-stage query row h into qs[c][w] (waited in first stage_row)
  {
    const float* rowp = query + nbase + (size_t)h * WW;
    for (int i = tid; i < CCH * (WW / 4); i += 256) {
      const int c = i >> 4, wq = (i & 15) * 4;
      const unsigned ldst = (unsigned)(uintptr_t)&qs[c * QP + wq];
      const float* g = rowp + (size_t)c * PP + wq;
      asm volatile("global_load_async_to_lds_b128 %0, %1, off" ::"v"(ldst),
                   "v"(g)
                   : "memory");
    }
  }

  // wave -> (M-tile of 16 pixels, N-tile of 16 band columns) for sim GEMM
  const int mt = wv >> 1;
  const int nt = wv & 1;
  const int w0 = mt * 16;
  const int n0 = w0 + nt * 16;

  // phase 1: sim via banded WMMA, one dy (shifted key row) at a time
  for (int dy = 0; dy < LWIN; ++dy) {
    __syncthreads();
    stage_row_issue(key, nbase, h + dy - PADW, rb, tid);
    async_wait0();
    __syncthreads();

    v8f acc = {};
#pragma unroll
    for (int kk = 0; kk < CCH / 4; ++kk) {
      const int k = kk * 4;
      v2f a, b;
      a.x = qs[(k + ks) * QP + w0 + lm];        // A = Q^T (M=w, K=c)
      a.y = qs[(k + ks + 1) * QP + w0 + lm];
      b.x = rb[(k + ks) * RP + n0 + lm];        // B = key row (K=c, N=col)
      b.y = rb[(k + ks + 1) * RP + n0 + lm];
      acc = __builtin_amdgcn_wmma_f32_16x16x4_f32(false, a, false, b, (short)0,
                                                  acc, false, false);
    }
    // scatter band entries: dx = col - pixel in [0, 8]
#pragma unroll
    for (int r = 0; r < 8; ++r) {
      const int row = w0 + r + hi * 8;
      const int dx  = (n0 + lm) - row;
      if (0 <= dx && dx <= 8) sim[row * SP + dy * LWIN + dx] = acc[r];
    }
  }
  __syncthreads();

  // phase 2: softmax over the 81 window scores (threads 0..63, one per pixel)
  if (tid < WW) {
    float m = -1e30f;
    for (int k = 0; k < KWIN; ++k) m = fmaxf(m, sim[tid * SP + k]);
    float s = 0.0f;
    for (int k = 0; k < KWIN; ++k) s += __expf(sim[tid * SP + k] - m);
    const float inv = 1.0f / s;
    for (int k = 0; k < KWIN; ++k)
      sim[tid * SP + k] = __expf(sim[tid * SP + k] - m) * inv;
  }

  // phase 3: weighting as banded WMMA GEMM.
  // out[c,w] += sum_{w'} ft_row[c,w'] * Wb[w',w],
  // Wb[w',w] = weight[w, dy*9 + (w'-w)] for w'-w in [0,8], else 0.
  // Wave wv owns channel tile c0 = wv*16 and all four 16-pixel tiles.
  const int c0 = wv * 16;
  v8f oacc[4];
  {
    v8f z = {};
#pragma unroll
    for (int t = 0; t < 4; ++t) oacc[t] = z;
  }

  for (int dy = 0; dy < LWIN; ++dy) {
    __syncthreads();
    stage_row_issue(ft, nbase, h + dy - PADW, rb, tid);  // async in flight...
    // ...overlap: build the banded weight matrix from softmax weights
    for (int i = tid; i < WCOLS * WW; i += 256) {
      const int wp = i >> 6;   // padded column w' (0..71)
      const int w  = i & 63;   // pixel
      const int dx = wp - w;
      wb[wp * WBP + w] =
          (0 <= dx && dx <= 8) ? sim[w * SP + dy * LWIN + dx] : 0.0f;
    }
    async_wait0();
    __syncthreads();

#pragma unroll
    for (int t = 0; t < 4; ++t) {
      const int wt0 = t * 16;
      v8f a8 = oacc[t];
      // band rows w' in [wt0, wt0+24) -> 6 k-steps of 4
#pragma unroll
      for (int kk = 0; kk < 6; ++kk) {
        const int k = kk * 4;
        v2f a, b;
        a.x = rb[(c0 + lm) * RP + wt0 + k + ks];        // A = ft row (M=c, K=w')
        a.y = rb[(c0 + lm) * RP + wt0 + k + ks + 1];
        b.x = wb[(wt0 + k + ks) * WBP + wt0 + lm];      // B = Wb (K=w', N=w)
        b.y = wb[(wt0 + k + ks + 1) * WBP + wt0 + lm];
        a8 = __builtin_amdgcn_wmma_f32_16x16x4_f32(false, a, false, b,
                                                   (short)0, a8, false, false);
      }
      oacc[t] = a8;
    }
  }

  // write D tiles: VGPR r -> channel c0 + r (+8 for upper half-wave), pixel lm
#pragma unroll
  for (int t = 0; t < 4; ++t) {
    const int c = c0 + hi * 8;
#pragma unroll
    for (int r = 0; r < 8; ++r)
      out[nbase + (size_t)(c + r) * PP + h * WW + t * 16 + lm] = oacc[t][r];
  }
}

// ---------------------------------------------------------------------------
extern "C" void kernel_launch(void* const* d_in, const int* in_sizes, int n_in,
                              void* d_out, int out_size, void* d_ws,
                              size_t ws_size, hipStream_t stream) {
  const float* ft = (const float*)d_in[0];
  const float* fk = (const float*)d_in[1];
  const float* Wm = (const float*)d_in[2];
  float* out = (float*)d_out;

  // workspace: key (2*128*4096 f32) then query (same) = 8 MB
  float* key   = (float*)d_ws;
  float* query = key + (size_t)2 * CCH * PP;

  ila_proj_gemm_f32<<<dim3(PP / 16, 2, 2), 256, 0, stream>>>(Wm, ft, fk, key,
                                                             query);

  const size_t lds =
      (size_t)(CCH * QP + CCH * RP + WW * SP + WCOLS * WBP) * sizeof(float);
  ila_attn_f32<<<dim3(HH, 2), 256, lds, stream>>>(ft, key, query, out);
}